// GCNHealingAgent_9096740733199
// MI455X (gfx1250) — compile-verified
//
#include <hip/hip_runtime.h>
#include <hip/hip_bf16.h>
#include <math.h>

#define GN 50000
#define GE 800000
#define GD 64

typedef __attribute__((ext_vector_type(2))) float v2f;
typedef __attribute__((ext_vector_type(8))) float v8f;

__device__ __forceinline__ v8f v8f_zero() {
    v8f z;
#pragma unroll
    for (int i = 0; i < 8; ++i) z[i] = 0.0f;
    return z;
}

// D = A(16x4 f32) * B(4x16 f32) + C(16x16 f32), wave32, full-precision WMMA.
__device__ __forceinline__ v8f wmma_f32(v2f a, v2f b, v8f c) {
    return __builtin_amdgcn_wmma_f32_16x16x4_f32(
        /*neg_a=*/false, a, /*neg_b=*/false, b,
        /*c_mod=*/(short)0, c, /*reuse_a=*/false, /*reuse_b=*/false);
}

// ---------------------------------------------------------------- utilities
__global__ void k_fill(float* __restrict__ p, float v, int n) {
    int i = blockIdx.x * blockDim.x + threadIdx.x;
    if (i < n) p[i] = v;
}

__global__ void k_degree(const int* __restrict__ dst, float* __restrict__ deg, int e) {
    int i = blockIdx.x * blockDim.x + threadIdx.x;
    if (i < e) atomicAdd(&deg[dst[i]], 1.0f);
}

__global__ void k_rsqrt_inplace(float* __restrict__ p, int n) {
    int i = blockIdx.x * blockDim.x + threadIdx.x;
    if (i < n) p[i] = rsqrtf(p[i]);
}

// ------------------------------------------------- h = x @ W_emb + b  (K=16)
__global__ void k_emb(const float* __restrict__ x, const float* __restrict__ W,
                      const float* __restrict__ bias, float* __restrict__ h, int n_tiles) {
    __shared__ float sW[16 * 64];
    for (int i = threadIdx.x; i < 16 * 64; i += blockDim.x) sW[i] = W[i];
    __syncthreads();

    int wave = blockIdx.x * (blockDim.x >> 5) + (threadIdx.x >> 5);
    int tile = wave < n_tiles ? wave : n_tiles - 1;   // keep EXEC all-ones
    int lane = threadIdx.x & 31;
    int m    = lane & 15;
    int kb   = (lane >> 4) << 1;                      // 0 or 2 (K half-split)
    int row0 = tile * 16;

    v8f c[4];
#pragma unroll
    for (int i = 0; i < 4; ++i) c[i] = v8f_zero();

#pragma unroll
    for (int kk = 0; kk < 4; ++kk) {
        int k0 = kk * 4 + kb;
        v2f a;
        a.x = x[(row0 + m) * 16 + k0];
        a.y = x[(row0 + m) * 16 + k0 + 1];
#pragma unroll
        for (int nt = 0; nt < 4; ++nt) {
            v2f b;
            b.x = sW[k0 * 64 + nt * 16 + m];
            b.y = sW[(k0 + 1) * 64 + nt * 16 + m];
            c[nt] = wmma_f32(a, b, c[nt]);
        }
    }

    int rbase = (lane < 16) ? 0 : 8;
#pragma unroll
    for (int nt = 0; nt < 4; ++nt)
#pragma unroll
        for (int r = 0; r < 8; ++r) {
            int row = row0 + rbase + r;
            int col = nt * 16 + m;
            h[row * 64 + col] = c[nt][r] + bias[col];
        }
}

// ------------------- hw = h @ Wc ; agg = hw * dinv^2 (self loop)  (K=64)
__global__ void k_conv_gemm(const float* __restrict__ h, const float* __restrict__ W,
                            const float* __restrict__ dinv, float* __restrict__ hw,
                            float* __restrict__ agg, int n_tiles) {
    __shared__ float sW[64 * 64];
    for (int i = threadIdx.x; i < 64 * 64; i += blockDim.x) sW[i] = W[i];
    __syncthreads();

    int wave = blockIdx.x * (blockDim.x >> 5) + (threadIdx.x >> 5);
    int tile = wave < n_tiles ? wave : n_tiles - 1;
    int lane = threadIdx.x & 31;
    int m    = lane & 15;
    int kb   = (lane >> 4) << 1;
    int row0 = tile * 16;

    v8f c[4];
#pragma unroll
    for (int i = 0; i < 4; ++i) c[i] = v8f_zero();

#pragma unroll
    for (int kk = 0; kk < 16; ++kk) {
        int k0 = kk * 4 + kb;
        v2f a;
        a.x = h[(row0 + m) * 64 + k0];
        a.y = h[(row0 + m) * 64 + k0 + 1];
#pragma unroll
        for (int nt = 0; nt < 4; ++nt) {
            v2f b;
            b.x = sW[k0 * 64 + nt * 16 + m];
            b.y = sW[(k0 + 1) * 64 + nt * 16 + m];
            c[nt] = wmma_f32(a, b, c[nt]);
        }
    }

    int rbase = (lane < 16) ? 0 : 8;
#pragma unroll
    for (int nt = 0; nt < 4; ++nt)
#pragma unroll
        for (int r = 0; r < 8; ++r) {
            int row = row0 + rbase + r;
            int col = nt * 16 + m;
            float v = c[nt][r];
            float di = dinv[row];
            hw[row * 64 + col]  = v;
            agg[row * 64 + col] = v * di * di;   // self-loop term
        }
}

// ----------------- edge scatter: agg[dst] += hw[src] * dinv[src]*dinv[dst]
// 64 lanes (2 waves) per edge: lane c handles component c -> coalesced
// gathers and coalesced global_atomic_add_f32 bursts (L2-resident).
__global__ void k_scatter(const int* __restrict__ src, const int* __restrict__ dst,
                          const float* __restrict__ dinv, const float* __restrict__ hw,
                          float* __restrict__ agg, int total) {
    int idx = blockIdx.x * blockDim.x + threadIdx.x;
    if (idx >= total) return;
    int e = idx >> 6;
    int c = idx & 63;
    int s = src[e];
    int d = dst[e];
    float nrm = dinv[s] * dinv[d];
    atomicAdd(&agg[d * 64 + c], hw[s * 64 + c] * nrm);
}

// --------------------------------- h = h + relu(agg + b)
__global__ void k_post(float* __restrict__ h, const float* __restrict__ agg,
                       const float* __restrict__ bias, int total) {
    int idx = blockIdx.x * blockDim.x + threadIdx.x;
    if (idx >= total) return;
    float v = agg[idx] + bias[idx & 63];
    h[idx] += fmaxf(v, 0.0f);
}

// ---- node head: t = relu(h@Wn1+b1); np = t@Wn2+b2 ; split into tl / pd
__global__ void k_node_head(const float* __restrict__ h,
                            const float* __restrict__ Wn1, const float* __restrict__ bn1,
                            const float* __restrict__ Wn2, const float* __restrict__ bn2,
                            float* __restrict__ tl, float* __restrict__ pd, int n_tiles) {
    __shared__ float sW1[64 * 64];
    __shared__ float sW2[64 * 16];     // Wn2 zero-padded 13 -> 16 cols
    __shared__ float sT[8][16 * 64];   // per-wave relu'd intermediate tile

    for (int i = threadIdx.x; i < 64 * 64; i += blockDim.x) sW1[i] = Wn1[i];
    for (int i = threadIdx.x; i < 64 * 16; i += blockDim.x) {
        int k = i >> 4, n = i & 15;
        sW2[i] = (n < 13) ? Wn2[k * 13 + n] : 0.0f;
    }
    __syncthreads();

    int wid  = threadIdx.x >> 5;
    int wave = blockIdx.x * (blockDim.x >> 5) + wid;
    int tile = wave < n_tiles ? wave : n_tiles - 1;
    int lane = threadIdx.x & 31;
    int m    = lane & 15;
    int kb   = (lane >> 4) << 1;
    int row0 = tile * 16;
    int rbase = (lane < 16) ? 0 : 8;

    // GEMM1: 16x64 tile of relu(h @ Wn1 + bn1)
    v8f c[4];
#pragma unroll
    for (int i = 0; i < 4; ++i) c[i] = v8f_zero();
#pragma unroll
    for (int kk = 0; kk < 16; ++kk) {
        int k0 = kk * 4 + kb;
        v2f a;
        a.x = h[(row0 + m) * 64 + k0];
        a.y = h[(row0 + m) * 64 + k0 + 1];
#pragma unroll
        for (int nt = 0; nt < 4; ++nt) {
            v2f b;
            b.x = sW1[k0 * 64 + nt * 16 + m];
            b.y = sW1[(k0 + 1) * 64 + nt * 16 + m];
            c[nt] = wmma_f32(a, b, c[nt]);
        }
    }
#pragma unroll
    for (int nt = 0; nt < 4; ++nt)
#pragma unroll
        for (int r = 0; r < 8; ++r) {
            int rl  = rbase + r;
            int col = nt * 16 + m;
            sT[wid][rl * 64 + col] = fmaxf(c[nt][r] + bn1[col], 0.0f);
        }
    __syncthreads();   // all waves reach here (tiles clamped, no early exit)

    // GEMM2: (16x64) @ (64x16-padded) ; A-fragments gathered from LDS tile
    v8f c2 = v8f_zero();
#pragma unroll
    for (int kk = 0; kk < 16; ++kk) {
        int k0 = kk * 4 + kb;
        v2f a;
        a.x = sT[wid][m * 64 + k0];
        a.y = sT[wid][m * 64 + k0 + 1];
        v2f b;
        b.x = sW2[k0 * 16 + m];
        b.y = sW2[(k0 + 1) * 16 + m];
        c2 = wmma_f32(a, b, c2);
    }
#pragma unroll
    for (int r = 0; r < 8; ++r) {
        int row = row0 + rbase + r;
        int col = m;
        float v = c2[r] + bn2[col < 13 ? col : 0];
        if (col < 10)      tl[row * 10 + col] = v;
        else if (col < 13) pd[row * 3 + (col - 10)] = v;
    }
}

// ---------------- column-sum partials for the mean pool (stride % 64 == 0)
__global__ void k_rowmean_partial(const float* __restrict__ h, float* __restrict__ acc,
                                  int total) {
    __shared__ float red[256];
    int stride = gridDim.x * blockDim.x;   // multiple of 64 -> column stays fixed
    float s = 0.0f;
    for (int idx = blockIdx.x * blockDim.x + threadIdx.x; idx < total; idx += stride)
        s += h[idx];
    red[threadIdx.x] = s;
    __syncthreads();
    if (threadIdx.x < 64) {
        float t = red[threadIdx.x] + red[threadIdx.x + 64] +
                  red[threadIdx.x + 128] + red[threadIdx.x + 192];
        atomicAdd(&acc[threadIdx.x], t);
    }
}

// -------- single-wave global head: sigmoid(relu(g@Wg1+b1)@Wg2+b2)
__global__ void k_global_head(const float* __restrict__ acc,
                              const float* __restrict__ Wg1, const float* __restrict__ bg1,
                              const float* __restrict__ Wg2, const float* __restrict__ bg2,
                              float* __restrict__ out, float invN) {
    int j = threadIdx.x;   // 32 lanes, one per hidden unit
    float hs = bg1[j];
#pragma unroll 8
    for (int k = 0; k < 64; ++k) hs += (acc[k] * invN) * Wg1[k * 32 + j];
    float hv = fmaxf(hs, 0.0f) * Wg2[j];
#pragma unroll
    for (int off = 16; off > 0; off >>= 1) hv += __shfl_down(hv, off, 32);
    if (j == 0) out[0] = 1.0f / (1.0f + expf(-(hv + bg2[0])));
}

extern "C" void kernel_launch(void* const* d_in, const int* in_sizes, int n_in,
                              void* d_out, int out_size, void* d_ws, size_t ws_size,
                              hipStream_t stream) {
    const float* x     = (const float*)d_in[0];
    const int*   ei    = (const int*)  d_in[1];
    const float* W_emb = (const float*)d_in[2];
    const float* b_emb = (const float*)d_in[3];
    const float* Wc[3] = {(const float*)d_in[4], (const float*)d_in[6], (const float*)d_in[8]};
    const float* bc[3] = {(const float*)d_in[5], (const float*)d_in[7], (const float*)d_in[9]};
    const float* Wn1   = (const float*)d_in[10];
    const float* bn1   = (const float*)d_in[11];
    const float* Wn2   = (const float*)d_in[12];
    const float* bn2   = (const float*)d_in[13];
    const float* Wg1   = (const float*)d_in[14];
    const float* bg1   = (const float*)d_in[15];
    const float* Wg2   = (const float*)d_in[16];
    const float* bg2   = (const float*)d_in[17];

    const int* src = ei;
    const int* dst = ei + GE;

    float* h  = (float*)d_out;                      // final h lives in d_out
    float* tl = h  + (size_t)GN * 64;               // type_logits (N,10)
    float* pd = tl + (size_t)GN * 10;               // pos_delta   (N,3)
    float* gp = pd + (size_t)GN * 3;                // global_pred (1)

    float* ws   = (float*)d_ws;
    float* dinv = ws;                               // N   (deg -> rsqrt in place)
    float* gacc = ws + GN;                          // 64
    float* hw   = ws + GN + 64;                     // N*64
    float* agg  = hw + (size_t)GN * 64;             // N*64

    // degree / dinv
    k_fill<<<(GN + 255) / 256, 256, 0, stream>>>(dinv, 1.0f, GN);
    k_degree<<<(GE + 255) / 256, 256, 0, stream>>>(dst, dinv, GE);
    k_rsqrt_inplace<<<(GN + 255) / 256, 256, 0, stream>>>(dinv, GN);

    const int n_tiles = GN / 16;                    // 3125 (exact)
    const int gw      = (n_tiles + 7) / 8;          // 8 waves / block

    // embedding
    k_emb<<<gw, 256, 0, stream>>>(x, W_emb, b_emb, h, n_tiles);

    // 3 GCN layers with residual relu
    for (int l = 0; l < 3; ++l) {
        k_conv_gemm<<<gw, 256, 0, stream>>>(h, Wc[l], dinv, hw, agg, n_tiles);
        k_scatter<<<(GE * 64) / 256, 256, 0, stream>>>(src, dst, dinv, hw, agg, GE * 64);
        k_post<<<(GN * 64) / 256, 256, 0, stream>>>(h, agg, bc[l], GN * 64);
    }

    // node head
    k_node_head<<<gw, 256, 0, stream>>>(h, Wn1, bn1, Wn2, bn2, tl, pd, n_tiles);

    // global head
    k_fill<<<1, 64, 0, stream>>>(gacc, 0.0f, 64);
    k_rowmean_partial<<<512, 256, 0, stream>>>(h, gacc, GN * 64);
    k_global_head<<<1, 32, 0, stream>>>(gacc, Wg1, bg1, Wg2, bg2, gp, 1.0f / (float)GN);
}